// RMSNorm_2886218023291
// MI455X (gfx1250) — compile-verified
//
#include <hip/hip_runtime.h>
#include <cstdint>
#include <cstddef>

// Fused residual-add + RMSNorm for MI455X (gfx1250, wave32).
// Memory-bound: ~1.0 GB of traffic -> ~44us floor at 23.3 TB/s.
// CDNA5 path: global_load_async_to_lds_b128 (NT, streaming) staging with
// pipelined s_wait_asynccnt, non-temporal stores for write-once outputs.

#define N_COLS 4096
#define BLOCK  256
#define WAVES  (BLOCK / 32)        // 8 waves of 32
#define CHUNK  (N_COLS / WAVES)    // 512 floats per wave
#define EPS_RMS 1e-6f

typedef float v4f __attribute__((ext_vector_type(4)));

// One async b128 op moves 32 lanes * 16B = 128 floats into LDS.
// NT hint: x/residual are read exactly once -> don't displace L2 (192MB)
// working lines (weight is the only reused input).
#define ASYNC_LOAD_NT(ldsaddr, byteoff, base)                                  \
    asm volatile("global_load_async_to_lds_b128 %0, %1, %2 th:TH_LOAD_NT"     \
                 :                                                             \
                 : "v"(ldsaddr), "v"(byteoff), "s"(base)                       \
                 : "memory")

__global__ __launch_bounds__(BLOCK) void fused_add_rmsnorm_kernel(
    const float* __restrict__ x,
    const float* __restrict__ r,
    const float* __restrict__ w,
    float* __restrict__ out,
    float* __restrict__ hout)
{
    __shared__ __align__(16) float ldsX[N_COLS];
    __shared__ __align__(16) float ldsR[N_COLS];
    __shared__ float red[WAVES];

    const int row  = blockIdx.x;
    const int tid  = threadIdx.x;
    const int wave = tid >> 5;
    const int lane = tid & 31;

    const size_t rowBase = (size_t)row * N_COLS;
    const unsigned long long xrow = (unsigned long long)(x + rowBase);
    const unsigned long long rrow = (unsigned long long)(r + rowBase);

    // --- Stage this wave's 512-float chunk of x and residual into LDS ---
    // Issue order: x0,r0,x1,r1,x2,r2,x3,r3  (8 outstanding on ASYNCcnt).
#pragma unroll
    for (int i = 0; i < 4; ++i) {
        const int idx = wave * CHUNK + i * 128 + lane * 4;   // float index
        const uint32_t boff = (uint32_t)idx * 4u;            // byte offset in row
        const uint32_t lx = (uint32_t)(uintptr_t)(const void*)&ldsX[idx];
        const uint32_t lr = (uint32_t)(uintptr_t)(const void*)&ldsR[idx];
        ASYNC_LOAD_NT(lx, boff, xrow);
        ASYNC_LOAD_NT(lr, boff, rrow);
    }

    // --- h = x + r, partial sum of squares; stream h to global (NT) ---
    // Async loads retire in order, so chunk j is ready once
    // ASYNCcnt <= 6 - 2j.  Pipeline compute against remaining fetches.
    v4f h[4];
    float sumsq = 0.f;

#define CONSUME_CHUNK(J, WAITN)                                                \
    do {                                                                       \
        asm volatile("s_wait_asynccnt " #WAITN ::: "memory");                  \
        const int idx = wave * CHUNK + (J) * 128 + lane * 4;                   \
        v4f a = *(const v4f*)&ldsX[idx];                                       \
        v4f b = *(const v4f*)&ldsR[idx];                                       \
        v4f hv = a + b;                                                        \
        h[J] = hv;                                                             \
        sumsq += hv.x * hv.x + hv.y * hv.y + hv.z * hv.z + hv.w * hv.w;        \
        __builtin_nontemporal_store(hv, (v4f*)(hout + rowBase + idx));         \
    } while (0)

    CONSUME_CHUNK(0, 6);
    CONSUME_CHUNK(1, 4);
    CONSUME_CHUNK(2, 2);
    CONSUME_CHUNK(3, 0);
#undef CONSUME_CHUNK

    // --- Cross-lane (wave32) + cross-wave reduction of sum of squares ---
#pragma unroll
    for (int off = 16; off > 0; off >>= 1)
        sumsq += __shfl_xor(sumsq, off, 32);
    if (lane == 0) red[wave] = sumsq;
    __syncthreads();

    float total = 0.f;
#pragma unroll
    for (int k = 0; k < WAVES; ++k)
        total += red[k];

    const float scale = rsqrtf(total * (1.0f / (float)N_COLS) + EPS_RMS);

    // --- out = h * scale * weight; weight stays cache-resident (RT loads) ---
#pragma unroll
    for (int j = 0; j < 4; ++j) {
        const int idx = wave * CHUNK + j * 128 + lane * 4;
        v4f wv = *(const v4f*)(w + idx);
        v4f o  = h[j] * scale * wv;
        __builtin_nontemporal_store(o, (v4f*)(out + rowBase + idx));
    }
}

extern "C" void kernel_launch(void* const* d_in, const int* in_sizes, int n_in,
                              void* d_out, int out_size, void* d_ws, size_t ws_size,
                              hipStream_t stream) {
    (void)n_in; (void)out_size; (void)d_ws; (void)ws_size;
    const float* x = (const float*)d_in[0];
    const float* r = (const float*)d_in[1];
    const float* w = (const float*)d_in[2];

    const int M = in_sizes[0] / N_COLS;          // 16384 rows
    float* out  = (float*)d_out;                 // first output: normalized
    float* hout = out + (size_t)M * N_COLS;      // second output: residual sum

    fused_add_rmsnorm_kernel<<<M, BLOCK, 0, stream>>>(x, r, w, out, hout);
}